// WeightAsyncHierGroupMultiLabelCE_12128987644156
// MI455X (gfx1250) — compile-verified
//
#include <hip/hip_runtime.h>
#include <float.h>

// Problem constants (from reference)
#define NB   4
#define NC   19
#define NPIX (512*512)
#define NS   2048
#define NSSM 8192

// Workspace layout in 4-byte words
#define N_MAXV (NB*NS*NC)          // f32-as-int max of weak softmax per (n,seg,c)
#define W_IDX  (N_MAXV)
#define N_IDX  (NB*NS*NC)          // argmax pixel index per (n,seg,c)
#define W_SUM  (W_IDX + N_IDX)
#define N_SUM  (NB*NSSM*NC)        // nll segment sums per (n,small,c)
#define W_CNT  (W_SUM + N_SUM)
#define N_CNT  (NB*NSSM)           // masked-pixel counts per (n,small)
#define W_ACC  (W_CNT + N_CNT)     // acc[0]=loss, acc[1]=nv
#define TOTAL_WS (W_ACC + 2)

typedef float v2f __attribute__((ext_vector_type(2)));
typedef float v8f __attribute__((ext_vector_type(8)));

// ---------------- init ----------------
__global__ void k_init(int* __restrict__ ws) {
  int i = blockIdx.x * blockDim.x + threadIdx.x;
  if (i >= TOTAL_WS) return;
  int v = 0;                                   // 0 bits == 0.0f: identity for positive-float max, zero for sums
  if (i >= W_IDX && i < W_IDX + N_IDX) v = NPIX;  // segment_min identity
  ws[i] = v;
}

// ---------------- softmax over 19 channels (channel-major layout) ----------------
__device__ __forceinline__ void softmax19(const float* __restrict__ base, int p, float* o) {
  float m = -FLT_MAX;
  #pragma unroll
  for (int c = 0; c < NC; ++c) { o[c] = base[(size_t)c * NPIX + p]; m = fmaxf(m, o[c]); }
  float s = 0.f;
  #pragma unroll
  for (int c = 0; c < NC; ++c) { o[c] = __expf(o[c] - m); s += o[c]; }
  float inv = 1.f / s;
  #pragma unroll
  for (int c = 0; c < NC; ++c) o[c] *= inv;
}

// ---------------- weak pass 1: per-(seg,c) max of softmax ----------------
__global__ void k_weak_max(const float* __restrict__ inw,
                           const unsigned char* __restrict__ maskw,
                           const int* __restrict__ segw,
                           int* __restrict__ maxv) {
  int t = blockIdx.x * blockDim.x + threadIdx.x;   // exact: NB*NPIX threads
  if (!maskw[t]) return;                            // masked pixels land in dropped segment S
  int n = t / NPIX, p = t - n * NPIX;
  int seg = segw[t];
  float o[NC];
  softmax19(inw + (size_t)n * NC * NPIX, p, o);
  int* row = maxv + ((size_t)n * NS + seg) * NC;
  #pragma unroll
  for (int c = 0; c < NC; ++c)
    atomicMax(&row[c], __float_as_int(o[c]));       // valid: softmax > 0
}

// ---------------- weak pass 2: min pixel index achieving the max ----------------
__global__ void k_weak_arg(const float* __restrict__ inw,
                           const unsigned char* __restrict__ maskw,
                           const int* __restrict__ segw,
                           const int* __restrict__ maxv,
                           int* __restrict__ idx) {
  int t = blockIdx.x * blockDim.x + threadIdx.x;
  if (!maskw[t]) return;
  int n = t / NPIX, p = t - n * NPIX;
  int seg = segw[t];
  float o[NC];
  softmax19(inw + (size_t)n * NC * NPIX, p, o);     // bit-identical recompute (L2-resident re-read)
  const int* mrow = maxv + ((size_t)n * NS + seg) * NC;
  int* irow = idx + ((size_t)n * NS + seg) * NC;
  #pragma unroll
  for (int c = 0; c < NC; ++c)
    if (__float_as_int(o[c]) == mrow[c]) atomicMin(&irow[c], p);
}

// ---------------- strong pass: nll segment sums + counts ----------------
__global__ void k_strong(const float* __restrict__ ins,
                         const unsigned char* __restrict__ masks,
                         const int* __restrict__ segsm,
                         float* __restrict__ sumpool,
                         int* __restrict__ counts) {
  int t = blockIdx.x * blockDim.x + threadIdx.x;
  if (!masks[t]) return;
  int n = t / NPIX, p = t - n * NPIX;
  int sm = segsm[t];
  float o[NC];
  softmax19(ins + (size_t)n * NC * NPIX, p, o);
  float* row = sumpool + ((size_t)n * NSSM + sm) * NC;
  #pragma unroll
  for (int c = 0; c < NC; ++c)
    atomicAdd(&row[c], -__logf(o[c] + 1e-8f));
  atomicAdd(&counts[(size_t)n * NSSM + sm], 1);
}

// ---------------- final gather + WMMA wave reduction ----------------
__global__ void k_pairs(const int* __restrict__ idx,
                        const float* __restrict__ targets,
                        const int* __restrict__ smallw,
                        const float* __restrict__ sumpool,
                        const int* __restrict__ counts,
                        float* __restrict__ acc) {
  int t = blockIdx.x * blockDim.x + threadIdx.x;   // exact: NB*NS*NC threads, no early exit (EXEC all-1s for WMMA)
  int c = t % NC;
  int s = (t / NC) % NS;
  int n = t / (NC * NS);
  float loss = 0.f, nv = 0.f;
  const int* irow = idx + ((size_t)n * NS + s) * NC;
  if (irow[0] < NPIX) {                             // has_pixel
    float trg = targets[((size_t)n * NS + s) * (NC + 1) + c];
    if (trg > 0.f) {
      int pix = irow[c];
      pix = pix < 0 ? 0 : (pix >= NPIX ? NPIX - 1 : pix);
      int sm = smallw[(size_t)n * NPIX + pix];
      loss = sumpool[((size_t)n * NSSM + sm) * NC + c];
      nv = (float)counts[(size_t)n * NSSM + sm];
    }
  }
#if defined(__AMDGCN__) && __has_builtin(__builtin_amdgcn_wmma_f32_16x16x4_f32)
  // Exact f32 cross-lane sum on the matrix unit:
  // A(16x4) = per-lane partials (VGPR0=loss, VGPR1=0), B(4x16)=ones.
  // D[m,*] = loss[lane m] + loss[lane m+16]; lanes 0-15 hold rows 0-7 in D[0..7],
  // lanes 16-31 hold rows 8-15 -> lane0 + lane16 together cover all 32 lanes.
  v2f b; b[0] = 1.0f; b[1] = 1.0f;
  v8f cz = {0.f, 0.f, 0.f, 0.f, 0.f, 0.f, 0.f, 0.f};
  v2f a; a[0] = loss; a[1] = 0.0f;
  v8f d = __builtin_amdgcn_wmma_f32_16x16x4_f32(false, a, false, b, (short)0, cz, false, false);
  float pl = d[0] + d[1] + d[2] + d[3] + d[4] + d[5] + d[6] + d[7];
  a[0] = nv;
  v8f d2 = __builtin_amdgcn_wmma_f32_16x16x4_f32(false, a, false, b, (short)0, cz, false, false);
  float pn = d2[0] + d2[1] + d2[2] + d2[3] + d2[4] + d2[5] + d2[6] + d2[7];
  int lane = threadIdx.x & 31;
  if (lane == 0 || lane == 16) {
    atomicAdd(&acc[0], pl);
    atomicAdd(&acc[1], pn);
  }
#else
  __shared__ float sl[256], sn[256];
  int tid = threadIdx.x;
  sl[tid] = loss; sn[tid] = nv;
  __syncthreads();
  for (int o = 128; o > 0; o >>= 1) {
    if (tid < o) { sl[tid] += sl[tid + o]; sn[tid] += sn[tid + o]; }
    __syncthreads();
  }
  if (tid == 0) { atomicAdd(&acc[0], sl[0]); atomicAdd(&acc[1], sn[0]); }
#endif
}

// ---------------- finalize ----------------
__global__ void k_final(const float* __restrict__ acc, float* __restrict__ out) {
  if (threadIdx.x == 0 && blockIdx.x == 0)
    out[0] = acc[0] / (1.0f + acc[1]);
}

extern "C" void kernel_launch(void* const* d_in, const int* in_sizes, int n_in,
                              void* d_out, int out_size, void* d_ws, size_t ws_size,
                              hipStream_t stream) {
  const float*         inputs            = (const float*)d_in[0];
  const float*         inputs_weak       = (const float*)d_in[1];
  const float*         targets           = (const float*)d_in[2];
  const unsigned char* spmasks           = (const unsigned char*)d_in[3];
  const unsigned char* spmasks_weak      = (const unsigned char*)d_in[4];
  /* d_in[5] = superpixels: unused by the reference */
  const int*           superpixels_weak  = (const int*)d_in[6];
  const int*           superpixel_smalls = (const int*)d_in[7];
  const int*           spx_smalls_weak   = (const int*)d_in[8];

  int*   ws      = (int*)d_ws;
  int*   maxv    = ws;
  int*   idx     = ws + W_IDX;
  float* sumpool = (float*)(ws + W_SUM);
  int*   counts  = ws + W_CNT;
  float* acc     = (float*)(ws + W_ACC);

  const int TB = 256;
  k_init<<<(TOTAL_WS + TB - 1) / TB, TB, 0, stream>>>(ws);

  const int pix_blocks = (NB * NPIX) / TB;  // 4096, exact
  k_weak_max<<<pix_blocks, TB, 0, stream>>>(inputs_weak, spmasks_weak, superpixels_weak, maxv);
  k_weak_arg<<<pix_blocks, TB, 0, stream>>>(inputs_weak, spmasks_weak, superpixels_weak, maxv, idx);
  k_strong  <<<pix_blocks, TB, 0, stream>>>(inputs, spmasks, superpixel_smalls, sumpool, counts);

  const int pair_blocks = (NB * NS * NC) / TB;  // 608, exact
  k_pairs<<<pair_blocks, TB, 0, stream>>>(idx, targets, spx_smalls_weak, sumpool, counts, acc);

  k_final<<<1, 32, 0, stream>>>(acc, (float*)d_out);
}